// LSSFPN_38903813767426
// MI455X (gfx1250) — compile-verified
//
#include <hip/hip_runtime.h>

typedef __attribute__((ext_vector_type(2))) float v2f;
typedef __attribute__((ext_vector_type(8))) float v8f;

#define N_CAM 6
#define DD    112
#define HH    16
#define WW    44
#define CC    32
#define BEV   128
#define CELLS (BEV * BEV)
#define HW    (HH * WW)

// ---------------------------------------------------------------------------
// Kernel 0: zero the BEV accumulator in workspace (2 MB), float4 stores.
// ---------------------------------------------------------------------------
__global__ void zero_ws_kernel(float4* __restrict__ ws) {
    int i = blockIdx.x * blockDim.x + threadIdx.x;
    ws[i] = make_float4(0.f, 0.f, 0.f, 0.f);
}

// ---------------------------------------------------------------------------
// Kernel 1: per-(camera, image-row, channel-half) tile:
//   LDS-staged softmax over D, WMMA rank-1 lift, coalesced atomic scatter.
// bev layout in workspace: [cell][CC] so a wave's 16 lanes hit 64B contiguous.
// s_cell is pre-scaled to (cell*CC + ch0) so the inner loop is add-only.
// ---------------------------------------------------------------------------
__global__ __launch_bounds__(256)
void lift_scatter_kernel(const float* __restrict__ hl,
                         const float* __restrict__ ctx,
                         const int*   __restrict__ geom,
                         float*       __restrict__ bev) {
    __shared__ float s_d[DD * WW];     // height logits -> softmax(depth)
    __shared__ int   s_cell[DD * WW];  // pre-scaled BEV offset per (d,w)
    __shared__ float s_ctx[WW * 16];   // 16 channels of context per pixel

    const int n   = blockIdx.x / HH;
    const int h   = blockIdx.x % HH;
    const int ch0 = blockIdx.y * 16;   // channel half: 0 or 16
    const int tid = threadIdx.x;

    // Stage height logits for this (n, h): element (d,w) -> hl[n, d, h, w]
    const int hlBase = n * (DD * HW) + h * WW;
    for (int i = tid; i < DD * WW; i += 256) {
        int d = i / WW, w = i % WW;
        s_d[i] = hl[hlBase + d * HW + w];
    }
    // Stage context channels [ch0, ch0+16): ctx[n, c, h, w] -> s_ctx[w*16 + (c-ch0)]
    const int ctxBase = n * (CC * HW) + h * WW;
    for (int i = tid; i < WW * 16; i += 256) {
        int w = i / 16, c = i % 16;
        s_ctx[i] = ctx[ctxBase + (ch0 + c) * HW + w];
    }
    // Stage pre-scaled BEV offsets: geom[0, n, d, h, w, {x,y}]
    const int gBase = n * (DD * HW) + h * WW;
    for (int i = tid; i < DD * WW; i += 256) {
        int d = i / WW, w = i % WW;
        int p = gBase + d * HW + w;
        int gx = geom[2 * p + 0];
        int gy = geom[2 * p + 1];
        s_cell[i] = (gy * BEV + gx) * CC + ch0;
    }
    __syncthreads();

    // Softmax over D for each pixel column w (threads 0..43, all-LDS).
    if (tid < WW) {
        float m = -3.402823466e38f;
        for (int d = 0; d < DD; ++d) m = fmaxf(m, s_d[d * WW + tid]);
        float s = 0.f;
        for (int d = 0; d < DD; ++d) {
            float e = __expf(s_d[d * WW + tid] - m);
            s_d[d * WW + tid] = e;
            s += e;
        }
        float inv = 1.f / s;
        for (int d = 0; d < DD; ++d) s_d[d * WW + tid] *= inv;
    }
    __syncthreads();

    const int wave = tid >> 5;
    const int lane = tid & 31;
    const int nIdx = lane & 15;          // channel within half (WMMA N index)
    const int mOff = (lane >> 4) << 3;   // lanes 16-31 hold rows M=8..15

    // Pixels striped over waves; 7 depth chunks of 16 per pixel.
    for (int w = wave; w < WW; w += 8) {
        // B(4x16 f32): lanes 0-15 VGPR0 = K=0 row (ctx); invariant over chunks.
        float bx = (lane < 16) ? s_ctx[w * 16 + lane] : 0.f;
        v2f B; B.x = bx; B.y = 0.f;

        for (int dc = 0; dc < DD / 16; ++dc) {
            const int dbase = dc * 16;

            // A(16x4 f32): lanes 0-15 VGPR0 = K=0 column (depth).
            float ax = (lane < 16) ? s_d[(dbase + lane) * WW + w] : 0.f;
            v2f A; A.x = ax; A.y = 0.f;
            v8f Cacc = {};
            // D[M][N] = depth[dbase+M] * ctx[ch0+N]  (rank-1 via K=0 only)
            v8f Dm = __builtin_amdgcn_wmma_f32_16x16x4_f32(
                false, A, false, B, (short)0, Cacc, false, false);

            // Batch all 8 cell lookups so the ds_loads pipeline under one
            // wait, then fire the atomic burst back-to-back.
            int off[8];
#pragma unroll
            for (int v = 0; v < 8; ++v)
                off[v] = s_cell[(dbase + mOff + v) * WW + w];
#pragma unroll
            for (int v = 0; v < 8; ++v)
                atomicAdd(&bev[off[v] + nIdx], Dm[v]);
        }
    }
}

// ---------------------------------------------------------------------------
// Kernel 2: [cell][C] -> [C][cell] with coalesced output writes.
// ---------------------------------------------------------------------------
__global__ void bev_transpose_kernel(const float* __restrict__ ws,
                                     float* __restrict__ out) {
    int i    = blockIdx.x * blockDim.x + threadIdx.x;  // over C*CELLS
    int c    = i / CELLS;
    int cell = i % CELLS;
    out[i] = ws[cell * CC + c];
}

// ---------------------------------------------------------------------------
extern "C" void kernel_launch(void* const* d_in, const int* in_sizes, int n_in,
                              void* d_out, int out_size, void* d_ws, size_t ws_size,
                              hipStream_t stream) {
    const float* hl   = (const float*)d_in[0];  // [6,112,16,44]
    const float* ctx  = (const float*)d_in[1];  // [6,32,16,44]
    const int*   geom = (const int*)d_in[2];    // [1,6,112,16,44,2]
    float* out = (float*)d_out;                 // [1,32,128,128]
    float* bev = (float*)d_ws;                  // [16384][32] accumulator

    // 524288 floats = 131072 float4 -> 512 blocks x 256
    zero_ws_kernel<<<(CELLS * CC / 4) / 256, 256, 0, stream>>>((float4*)d_ws);

    // 6*16 = 96 tiles x 2 channel halves = 192 workgroups
    lift_scatter_kernel<<<dim3(N_CAM * HH, 2), 256, 0, stream>>>(hl, ctx, geom, bev);

    // 524288 elements -> 2048 blocks x 256
    bev_transpose_kernel<<<(CELLS * CC) / 256, 256, 0, stream>>>(bev, out);
}